// SubglacialDrainageSystem_22084721836290
// MI455X (gfx1250) — compile-verified
//
#include <hip/hip_runtime.h>
#include <hip/hip_bf16.h>

typedef float v2f __attribute__((ext_vector_type(2)));
typedef float v8f __attribute__((ext_vector_type(8)));

#define LDW 65            // 64 + 1 pad: column reads hit distinct LDS banks
#define CG_ITERS 2000
#define SEC_PER_A 31556926.0f
#define K_COND 0.01f
#define HR_C 0.1f
#define LR_C 2.0f
#define A_CLOS 5e-25f

__device__ __forceinline__ v8f wmma4(v2f a, v2f b, v8f c) {
  // D = A(16x4,f32) * B(4x16,f32) + C(16x16,f32)  -> v_wmma_f32_16x16x4_f32
  return __builtin_amdgcn_wmma_f32_16x16x4_f32(false, a, false, b, (short)0, c,
                                               false, false);
}

// P-fragment is the A operand (east/west shifts: P * S^{T}/S).
// idx[j] = LDS index of A elem (row, k); pair (k, k+1) contiguous.
__device__ __forceinline__ v8f spmvA(const float* __restrict__ Pg,
                                     const int* idx, const v2f* bf) {
  v8f acc = {0.f,0.f,0.f,0.f,0.f,0.f,0.f,0.f};
#pragma unroll
  for (int j = 0; j < 8; ++j) {
    v2f a;
    a.x = Pg[idx[j]];
    a.y = Pg[idx[j] + 1];
    acc = wmma4(a, bf[j], acc);
  }
  return acc;
}

// P-fragment is the B operand (up/down shifts: S^{T}/S * P).
// idx[j] = LDS index of B elem (k, col); pair (k, k+1) is LDW apart.
__device__ __forceinline__ v8f spmvB(const float* __restrict__ Pg,
                                     const int* idx, const v2f* af) {
  v8f acc = {0.f,0.f,0.f,0.f,0.f,0.f,0.f,0.f};
#pragma unroll
  for (int j = 0; j < 8; ++j) {
    v2f b;
    b.x = Pg[idx[j]];
    b.y = Pg[idx[j] + LDW];
    acc = wmma4(af[j], b, acc);
  }
  return acc;
}

__device__ __forceinline__ float block_sum(float v, volatile float* red, int tid) {
  __syncthreads();                       // protect red from previous use
  for (int off = 16; off; off >>= 1) v += __shfl_xor(v, off, 32);
  if ((tid & 31) == 0) red[tid >> 5] = v;
  __syncthreads();
  if (tid == 0) {
    float s = 0.f;
    for (int i = 0; i < 16; ++i) s += red[i];
    red[16] = s;
  }
  __syncthreads();
  return red[16];
}

extern "C" __global__ __launch_bounds__(512, 1)
void SubglacialDrainageSystem_22084721836290_kernel(
    const float* __restrict__ base, const float* __restrict__ ovb,
    const float* __restrict__ melt, const float* __restrict__ sheet,
    const float* __restrict__ pot,  const float* __restrict__ sv,
    const float* __restrict__ llen, const int* __restrict__ lan,
    const int* __restrict__ inflow, const int* __restrict__ dtp,
    float* __restrict__ out) {
  __shared__ float Pg[64 * LDW];   // CG search direction (grid layout)
  __shared__ float G1[64 * LDW];   // sheet -> cE coefficient grid
  __shared__ float G2[64 * LDW];   // potential -> cD coefficient grid
  __shared__ float red[17];

  const int tid  = threadIdx.x;
  const int lane = tid & 31, wid = tid >> 5;
  const int lh = lane >> 4, lm = lane & 15;
  const int tr = wid >> 2, tc = wid & 3;
  const int trb = tr * 16, tcb = tc * 16;
  const int Cc = tcb + lm;                 // column of all 8 owned nodes

  // ---- stage input fields into LDS via async global->LDS DMA (ASYNCcnt) ----
  for (int i = tid * 4; i < 4096; i += 2048) {
    int r = i >> 6, c = i & 63;
    unsigned l1 = (unsigned)(size_t)&G1[r * LDW + c];
    unsigned l2 = (unsigned)(size_t)&G2[r * LDW + c];
    asm volatile("global_load_async_to_lds_b128 %0, %1, off"
                 :: "v"(l1), "v"(sheet + i) : "memory");
    asm volatile("global_load_async_to_lds_b128 %0, %1, off"
                 :: "v"(l2), "v"(pot + i) : "memory");
  }
  asm volatile("s_wait_asynccnt 0x0" ::: "memory");
  __syncthreads();

  // ---- precompute loop-invariant shift-matrix WMMA fragments + LDS indices ----
  // east  = P*S^T : blocks kb in {tc,   tc+1};  B[k][n] = (k   == n+1)
  // west  = P*S   : blocks kb in {tc-1, tc  };  B[k][n] = (n   == k+1)
  // up    = S^T*P : blocks kb in {tr-1, tr  };  A[m][k] = (m   == k+1)
  // down  = S*P   : blocks kb in {tr,   tr+1};  A[m][k] = (k   == m+1)
  // Out-of-range kb -> zero fragment (WMMA no-op); P index clamped in-bounds.
  v2f bE[8], bW[8], aU[8], aD[8];
  int iE[8], iW[8], iU[8], iD[8];
  const int rowb = (trb + lm) * LDW;       // A-frag row base (east/west)
  const int mrow = trb + lm;               // A-frag row index (up/down const)
#pragma unroll
  for (int j = 0; j < 8; ++j) {
    const int blk = j >> 2, ks = j & 3;
    {  // east
      int kb = tc + blk;
      float val = (kb <= 3) ? 1.f : 0.f;
      int k = kb * 16 + ks * 4 + 2 * lh;
      bE[j].x = (k     == Cc + 1) ? val : 0.f;
      bE[j].y = (k + 1 == Cc + 1) ? val : 0.f;
      int kc = k < 0 ? 0 : (k > 62 ? 62 : k);
      iE[j] = rowb + kc;
    }
    {  // west
      int kb = tc - 1 + blk;
      float val = (kb >= 0) ? 1.f : 0.f;
      int k = kb * 16 + ks * 4 + 2 * lh;
      bW[j].x = (Cc == k + 1) ? val : 0.f;
      bW[j].y = (Cc == k + 2) ? val : 0.f;
      int kc = k < 0 ? 0 : (k > 62 ? 62 : k);
      iW[j] = rowb + kc;
    }
    {  // up
      int kb = tr - 1 + blk;
      float val = (kb >= 0) ? 1.f : 0.f;
      int k = kb * 16 + ks * 4 + 2 * lh;
      aU[j].x = (mrow == k + 1) ? val : 0.f;
      aU[j].y = (mrow == k + 2) ? val : 0.f;
      int kc = k < 0 ? 0 : (k > 62 ? 62 : k);
      iU[j] = kc * LDW + Cc;
    }
    {  // down
      int kb = tr + blk;
      float val = (kb <= 3) ? 1.f : 0.f;
      int k = kb * 16 + ks * 4 + 2 * lh;
      aD[j].x = (k     == mrow + 1) ? val : 0.f;
      aD[j].y = (k + 1 == mrow + 1) ? val : 0.f;
      int kc = k < 0 ? 0 : (k > 62 ? 62 : k);
      iD[j] = kc * LDW + Cc;
    }
  }

  // ---- per-link coefficients (east link cE, down link cD per node) ----
  float cE[8], cD[8];
#pragma unroll
  for (int v = 0; v < 8; ++v) {
    int r = trb + v + 8 * lh;
    float s0 = G1[r * LDW + Cc], p0 = G2[r * LDW + Cc];
    cE[v] = 0.f;
    if (Cc < 63) {
      int l = r * 63 + Cc;
      float len = llen[l];
      float sh  = 0.5f * (s0 + G1[r * LDW + Cc + 1]);
      float dph = p0 - G2[r * LDW + Cc + 1];
      cE[v] = -K_COND * (sh * sqrtf(sqrtf(sh))) * len * rsqrtf(fabsf(dph) / len);
    }
    cD[v] = 0.f;
    if (r < 63) {
      int l = 4032 + r * 64 + Cc;
      float len = llen[l];
      float sh  = 0.5f * (s0 + G1[(r + 1) * LDW + Cc]);
      float dph = p0 - G2[(r + 1) * LDW + Cc];
      cD[v] = -K_COND * (sh * sqrtf(sqrtf(sh))) * len * rsqrtf(fabsf(dph) / len);
    }
  }
  __syncthreads();
#pragma unroll
  for (int v = 0; v < 8; ++v) {
    int r = trb + v + 8 * lh;
    G1[r * LDW + Cc] = cE[v];      // overwrite staging grids with coefficients
    G2[r * LDW + Cc] = cD[v];
  }
  __syncthreads();

  // ---- per-owned-node row data: coefficients, diag, Jacobi, forcing, x0 ----
  float cEo[8], cWo[8], cNo[8], cSo[8], dg[8], dinv[8], ff[8], xx[8];
#pragma unroll
  for (int v = 0; v < 8; ++v) {
    int r = trb + v + 8 * lh;
    int node = r * 64 + Cc;
    float e = cE[v];
    float w = (Cc > 0) ? G1[r * LDW + Cc - 1] : 0.f;
    float s = cD[v];
    float n = (r > 0) ? G2[(r - 1) * LDW + Cc] : 0.f;
    bool dir = (inflow[node] == 1);
    if (dir) { e = w = s = n = 0.f; }
    cEo[v] = e; cWo[v] = w; cSo[v] = s; cNo[v] = n;
    float d = -(e + w + s + n);
    dg[v]   = dir ? 0.f : d;
    dinv[v] = dir ? 0.f : 1.f / d;
    ff[v]   = dir ? (base[node] - ovb[node]) : melt[node];
    xx[v]   = dir ? ff[v] : 0.f;
    Pg[r * LDW + Cc] = xx[v];      // Pg := x0 for initial residual SpMV
  }
  __syncthreads();

  // ---- initial residual r0 = f - A*x0 (masked), z = D^-1 r, p = z ----
  float rr[8], pw[8];
  float rz;
  {
    v8f e = spmvA(Pg, iE, bE);
    v8f w = spmvA(Pg, iW, bW);
    v8f u = spmvB(Pg, iU, aU);
    v8f d = spmvB(Pg, iD, aD);
    float rzl = 0.f;
#pragma unroll
    for (int v = 0; v < 8; ++v) {
      float ap = dg[v] * xx[v] + cEo[v] * e[v] + cWo[v] * w[v] +
                 cNo[v] * u[v] + cSo[v] * d[v];
      rr[v] = ff[v] - ap;          // value at Dirichlet rows is never used
      float z = dinv[v] * rr[v];   // 0 at Dirichlet
      pw[v] = z;
      rzl += rr[v] * z;
    }
    rz = block_sum(rzl, red, tid);
#pragma unroll
    for (int v = 0; v < 8; ++v)
      Pg[(trb + v + 8 * lh) * LDW + Cc] = pw[v];
    __syncthreads();
  }

  // ---- Jacobi-preconditioned CG; stencil applied via f32 WMMA shifts ----
  for (int it = 0; it < CG_ITERS; ++it) {
    v8f e = spmvA(Pg, iE, bE);
    v8f w = spmvA(Pg, iW, bW);
    v8f u = spmvB(Pg, iU, aU);
    v8f d = spmvB(Pg, iD, aD);
    float ap[8], pApl = 0.f;
#pragma unroll
    for (int v = 0; v < 8; ++v) {
      ap[v] = dg[v] * pw[v] + cEo[v] * e[v] + cWo[v] * w[v] +
              cNo[v] * u[v] + cSo[v] * d[v];
      pApl += pw[v] * ap[v];
    }
    float pAp = block_sum(pApl, red, tid);
    float alpha = (pAp != 0.f) ? rz / pAp : 0.f;
    float rzl = 0.f;
#pragma unroll
    for (int v = 0; v < 8; ++v) {
      xx[v] += alpha * pw[v];
      rr[v] -= alpha * ap[v];
      rzl += rr[v] * (dinv[v] * rr[v]);
    }
    float rzn = block_sum(rzl, red, tid);
    float beta = (rz != 0.f) ? rzn / rz : 0.f;
    rz = rzn;
#pragma unroll
    for (int v = 0; v < 8; ++v) {
      pw[v] = dinv[v] * rr[v] + beta * pw[v];
      Pg[(trb + v + 8 * lh) * LDW + Cc] = pw[v];
    }
    __syncthreads();
  }

  // ---- outputs: new_potential, then closed-form sheet thickness update ----
  const float dtf = (float)dtp[0];
#pragma unroll
  for (int v = 0; v < 8; ++v) {
    int r = trb + v + 8 * lh;
    int node = r * 64 + Cc;
    out[node] = xx[v];
    float ssum = 0.f; int cnt = 0;
#pragma unroll
    for (int j = 0; j < 4; ++j) {
      int l = lan[node * 4 + j];
      if (l != -1) { ssum += sv[l]; ++cnt; }
    }
    int den = (cnt > 0) ? cnt : 1;
    float sliding = fabsf((ssum / SEC_PER_A) / (float)den);
    float P = base[node] - xx[v];
    float h0 = sheet[node];
    float nh = (h0 + dtf * sliding * HR_C / LR_C) /
               (1.0f + dtf * (sliding / LR_C + A_CLOS * P * P * P));
    out[4096 + node] = nh;
  }
}

extern "C" void kernel_launch(void* const* d_in, const int* in_sizes, int n_in,
                              void* d_out, int out_size, void* d_ws, size_t ws_size,
                              hipStream_t stream) {
  const float* base  = (const float*)d_in[0];
  const float* ovb   = (const float*)d_in[1];
  const float* melt  = (const float*)d_in[2];
  const float* sheet = (const float*)d_in[3];
  const float* pot   = (const float*)d_in[4];
  const float* sv    = (const float*)d_in[5];
  const float* llen  = (const float*)d_in[6];
  // d_in[7]=link_tail, d_in[8]=link_head unused (grid structure is analytic)
  const int*   lan    = (const int*)d_in[9];
  const int*   inflow = (const int*)d_in[10];
  const int*   dtp    = (const int*)d_in[11];
  hipLaunchKernelGGL(SubglacialDrainageSystem_22084721836290_kernel,
                     dim3(1), dim3(512), 0, stream,
                     base, ovb, melt, sheet, pot, sv, llen, lan, inflow, dtp,
                     (float*)d_out);
}